// LocalModule_7275674600087
// MI455X (gfx1250) — compile-verified
//
#include <hip/hip_runtime.h>

#define DD 128

typedef __attribute__((ext_vector_type(2))) float v2f;
typedef __attribute__((ext_vector_type(8))) float v8f;

// ---------------------------------------------------------------------------
// Zero the aggregation buffer + BN accumulators (must be re-zeroed every call)
// ---------------------------------------------------------------------------
__global__ void init_zero_kernel(float* __restrict__ p, long long n) {
  long long i = (long long)blockIdx.x * blockDim.x + threadIdx.x;
  if (i < n) p[i] = 0.0f;
}

// ---------------------------------------------------------------------------
// Edge phase: aggr[dst] += relu(x[src] + edge_attr)   (E x D elementwise)
// edge_attr stream (410MB) is the HBM-bound part; x/aggr are L2-resident.
// ---------------------------------------------------------------------------
__global__ void edge_msg_kernel(const float* __restrict__ x,
                                const int* __restrict__ src,
                                const int* __restrict__ dst,
                                const float* __restrict__ ea,
                                float* __restrict__ aggr,
                                long long total) {
  long long i = (long long)blockIdx.x * blockDim.x + threadIdx.x;
  if (i >= total) return;
  int e = (int)(i >> 7);       // D == 128
  int d = (int)(i & 127);
  float m = x[(long long)src[e] * DD + d] + ea[i];
  if (m > 0.0f) atomicAdd(&aggr[(long long)dst[e] * DD + d], m);
}

// ---------------------------------------------------------------------------
// h0 = x + aggr   (written into d_out; consumed + overwritten by MLP kernel)
// ---------------------------------------------------------------------------
__global__ void add_kernel(const float* __restrict__ a,
                           const float* __restrict__ b,
                           float* __restrict__ o, long long n) {
  long long i = (long long)blockIdx.x * blockDim.x + threadIdx.x;
  if (i < n) o[i] = a[i] + b[i];
}

// ---------------------------------------------------------------------------
// GIN MLP: h = x + (relu(h0 @ W1 + b1) @ W2 + b2), plus BN column sum/sumsq.
// One wave per 16-row tile; full 16x128 output strip in 8 v8f accumulators.
// V_WMMA_F32_16X16X4_F32, K stepped by 4 (32 steps x 8 N-tiles x 2 layers).
// Layer-1 result goes through a per-wave LDS strip to re-load in A-layout.
// ---------------------------------------------------------------------------
__global__ __launch_bounds__(128) void gin_mlp_kernel(
    const float* __restrict__ x,
    float* h,                       // h0 on input, final h on output (in-place)
    const float* __restrict__ W1, const float* __restrict__ b1,
    const float* __restrict__ W2, const float* __restrict__ b2,
    float* __restrict__ colsum, float* __restrict__ colsq,
    int nTiles) {
  __shared__ float sH1[4 * 16 * DD];   // 32 KB: 4 waves x (16 x 128) f32
  __shared__ float sSum[DD];
  __shared__ float sSq[DD];

  int tid = threadIdx.x;
  if (tid < DD) { sSum[tid] = 0.0f; sSq[tid] = 0.0f; }
  __syncthreads();

  int wave = tid >> 5;
  int lane = tid & 31;
  int tile = blockIdx.x * 4 + wave;
  bool active = tile < nTiles;
  if (!active) tile = nTiles - 1;      // clamp; keep all waves convergent

  int rowBase = tile * 16;
  int aM    = lane & 15;               // A frag: M row;  B/C frags: N column
  int kHalf = (lane >> 4) * 2;         // 0 for lanes 0-15, 2 for lanes 16-31
  int cM    = (lane >> 4) * 8;         // C frag: row offset of upper half

  const float* aRow = h + (long long)(rowBase + aM) * DD + kHalf;

  v8f acc[8];

  // ---- layer 1: relu(h0 @ W1 + b1) ----
  for (int t = 0; t < 8; ++t) {
    float bv = b1[t * 16 + aM];
    for (int r = 0; r < 8; ++r) acc[t][r] = bv;
  }
  for (int k0 = 0; k0 < DD; k0 += 4) {
    v2f a = *(const v2f*)(aRow + k0);
    const float* w = W1 + (long long)(k0 + kHalf) * DD + aM;
    for (int t = 0; t < 8; ++t) {
      v2f b;
      b.x = w[t * 16];
      b.y = w[t * 16 + DD];
      acc[t] = __builtin_amdgcn_wmma_f32_16x16x4_f32(
          false, a, false, b, (short)0, acc[t], false, false);
    }
  }

  // relu, spill to per-wave LDS strip in row-major [16 x 128]
  float* sh = &sH1[wave * 16 * DD];
  for (int t = 0; t < 8; ++t)
    for (int r = 0; r < 8; ++r) {
      float v = acc[t][r];
      sh[(cM + r) * DD + t * 16 + aM] = v > 0.0f ? v : 0.0f;
    }
  __syncthreads();

  // ---- layer 2: h1 @ W2 + b2 ----
  for (int t = 0; t < 8; ++t) {
    float bv = b2[t * 16 + aM];
    for (int r = 0; r < 8; ++r) acc[t][r] = bv;
  }
  const float* a2 = sh + aM * DD + kHalf;
  for (int k0 = 0; k0 < DD; k0 += 4) {
    v2f a = *(const v2f*)(a2 + k0);
    const float* w = W2 + (long long)(k0 + kHalf) * DD + aM;
    for (int t = 0; t < 8; ++t) {
      v2f b;
      b.x = w[t * 16];
      b.y = w[t * 16 + DD];
      acc[t] = __builtin_amdgcn_wmma_f32_16x16x4_f32(
          false, a, false, b, (short)0, acc[t], false, false);
    }
  }

  // residual + store + BN partial sums (block-level, then one global atomic)
  if (active) {
    for (int t = 0; t < 8; ++t) {
      int col = t * 16 + aM;
      float s = 0.0f, sq = 0.0f;
      for (int r = 0; r < 8; ++r) {
        long long idx = (long long)(rowBase + cM + r) * DD + col;
        float v = acc[t][r] + x[idx];
        h[idx] = v;
        s += v;
        sq += v * v;
      }
      atomicAdd(&sSum[col], s);   // ds_add_f32
      atomicAdd(&sSq[col], sq);
    }
  }
  __syncthreads();
  if (tid < DD) {
    atomicAdd(&colsum[tid], sSum[tid]);
    atomicAdd(&colsq[tid], sSq[tid]);
  }
}

// ---------------------------------------------------------------------------
// BN finalize: per-column scale/shift from sums (single 128-thread block)
// ---------------------------------------------------------------------------
__global__ void bn_stats_kernel(const float* __restrict__ colsum,
                                const float* __restrict__ colsq,
                                const float* __restrict__ bn_w,
                                const float* __restrict__ bn_b,
                                float* __restrict__ scale,
                                float* __restrict__ shift, float invN) {
  int d = threadIdx.x;
  float mean = colsum[d] * invN;
  float var = colsq[d] * invN - mean * mean;  // biased variance
  float sc = bn_w[d] * rsqrtf(var + 1e-5f);
  scale[d] = sc;
  shift[d] = bn_b[d] - mean * sc;
}

// ---------------------------------------------------------------------------
// BN apply, in place on d_out
// ---------------------------------------------------------------------------
__global__ void bn_apply_kernel(float* __restrict__ h,
                                const float* __restrict__ scale,
                                const float* __restrict__ shift, long long n) {
  long long i = (long long)blockIdx.x * blockDim.x + threadIdx.x;
  if (i < n) {
    int d = (int)(i & 127);
    h[i] = h[i] * scale[d] + shift[d];
  }
}

extern "C" void kernel_launch(void* const* d_in, const int* in_sizes, int n_in,
                              void* d_out, int out_size, void* d_ws, size_t ws_size,
                              hipStream_t stream) {
  const float* x    = (const float*)d_in[0];
  const int*   ei   = (const int*)d_in[1];
  const float* ea   = (const float*)d_in[2];
  const float* W1   = (const float*)d_in[3];
  const float* b1   = (const float*)d_in[4];
  const float* W2   = (const float*)d_in[5];
  const float* b2   = (const float*)d_in[6];
  const float* bn_w = (const float*)d_in[7];
  const float* bn_b = (const float*)d_in[8];

  long long ND = (long long)in_sizes[0];      // N * 128
  int N = (int)(ND / DD);
  long long E = (long long)in_sizes[1] / 2;
  const int* src = ei;                        // edge_index row 0
  const int* dst = ei + E;                    // edge_index row 1

  float* aggr   = (float*)d_ws;               // N*128 floats
  float* colsum = aggr + ND;                  // 128
  float* colsq  = colsum + DD;                // 128
  float* scale  = colsq + DD;                 // 128
  float* shift  = scale + DD;                 // 128

  float* h = (float*)d_out;                   // h0, then h, then normalized h

  // zero aggr + colsum + colsq (contiguous)
  long long zn = ND + 2 * DD;
  init_zero_kernel<<<(int)((zn + 255) / 256), 256, 0, stream>>>(aggr, zn);

  // edge scatter
  long long tot = E * DD;
  edge_msg_kernel<<<(int)((tot + 255) / 256), 256, 0, stream>>>(
      x, src, dst, ea, aggr, tot);

  // h0 = x + aggr
  add_kernel<<<(int)((ND + 255) / 256), 256, 0, stream>>>(x, aggr, h, ND);

  // WMMA MLP + residual + BN partial sums
  int nTiles = (N + 15) / 16;                 // 3125 (N divisible by 16)
  int blocks = (nTiles + 3) / 4;              // 4 waves / block
  gin_mlp_kernel<<<blocks, 128, 0, stream>>>(
      x, h, W1, b1, W2, b2, colsum, colsq, nTiles);

  // BN finalize + apply
  bn_stats_kernel<<<1, DD, 0, stream>>>(colsum, colsq, bn_w, bn_b, scale,
                                        shift, 1.0f / (float)N);
  bn_apply_kernel<<<(int)((ND + 255) / 256), 256, 0, stream>>>(h, scale, shift, ND);
}